// LightGCNConv_61675730371171
// MI455X (gfx1250) — compile-verified
//
#include <hip/hip_runtime.h>
#include <cstdint>

#define EMB_DIM 64
#define TPE     16                 // threads per edge: 16 x float4 = 64 floats
#define BLOCK   256
#define EPT     (BLOCK / TPE)      // edges per tile = 16
#define DEPTH   4                  // async pipeline depth (LDS ring slots)

// ---------------------------------------------------------------- zero accum
__global__ __launch_bounds__(256) void lgc_zero_kernel(float4* __restrict__ p, int n4) {
  int i = blockIdx.x * 256 + threadIdx.x;
  if (i < n4) p[i] = make_float4(0.f, 0.f, 0.f, 0.f);
}

// ------------------------------------------------- async gather-tile issuer
__device__ __forceinline__ void issue_tile(
    const float* __restrict__ emb,
    const int*   __restrict__ src,
    const int*   __restrict__ dst,
    const float* __restrict__ w,
    int nEdges, int strideEdges,
    int tile, int eoff, int sub, uint32_t ldsAddr,
    float& wv, int& dv, int& ev)
{
  int e  = tile * EPT + eoff;
  ev     = e;
  int ec = (e < nEdges) ? e : (nEdges - 1);

  // prefetch the index stream ahead (global_prefetch_b8)
  int pf = e + DEPTH * strideEdges;
  pf = (pf < nEdges) ? pf : (nEdges - 1);
  __builtin_prefetch(src + pf, 0, 0);

  int s = src[ec];
  dv    = dst[ec];
  wv    = w[ec];

  // per-lane async gather of 16B of the source row straight into LDS
  uint64_t ga = (uint64_t)(uintptr_t)emb
              + (uint64_t)(uint32_t)s * (uint64_t)(EMB_DIM * sizeof(float))
              + (uint64_t)(sub * 16);
  asm volatile("global_load_async_to_lds_b128 %0, %1, off"
               :
               : "v"(ldsAddr), "v"(ga)
               : "memory");
}

// -------------------------------------------- no-return HW f32 atomics (L2)
__device__ __forceinline__ void atomic_add4(float* base, float x, float y, float z, float ww) {
  uint64_t a = (uint64_t)(uintptr_t)base;
  asm volatile("global_atomic_add_f32 %0, %1, off"           : : "v"(a), "v"(x)  : "memory");
  asm volatile("global_atomic_add_f32 %0, %1, off offset:4"  : : "v"(a), "v"(y)  : "memory");
  asm volatile("global_atomic_add_f32 %0, %1, off offset:8"  : : "v"(a), "v"(z)  : "memory");
  asm volatile("global_atomic_add_f32 %0, %1, off offset:12" : : "v"(a), "v"(ww) : "memory");
}

// -------------------------------------------------- consume one ring slot
__device__ __forceinline__ void process_slot(
    const float4* __restrict__ shSlot,
    float wv, int dv, int ev,
    int nEdges, int sub, float* __restrict__ out)
{
  if (ev < nEdges) {
    float4 v = *shSlot;   // ds_load_b128; dscnt wait precedes the atomic uses
    float* o = out + (uint64_t)(uint32_t)dv * EMB_DIM + sub * 4;
    atomic_add4(o, v.x * wv, v.y * wv, v.z * wv, v.w * wv);
  }
}

// ------------------------------------------------------------- edge scatter
__global__ __launch_bounds__(BLOCK) void lgc_edge_kernel(
    const float* __restrict__ emb,
    const float* __restrict__ w,
    const int*   __restrict__ src,
    const int*   __restrict__ dst,
    float*       __restrict__ out,
    int nEdges, int nTiles)
{
  __shared__ float4 sh[DEPTH][BLOCK];

  const int tid  = threadIdx.x;
  const int sub  = tid & (TPE - 1);   // which float4 of the row
  const int eoff = tid >> 4;          // which edge within the tile
  const int grid = (int)gridDim.x;
  const int strideEdges = grid * EPT;

  const uint32_t lds0 = (uint32_t)(uintptr_t)(&sh[0][tid]);
  const uint32_t lds1 = (uint32_t)(uintptr_t)(&sh[1][tid]);
  const uint32_t lds2 = (uint32_t)(uintptr_t)(&sh[2][tid]);
  const uint32_t lds3 = (uint32_t)(uintptr_t)(&sh[3][tid]);

  const int t0 = (int)blockIdx.x;

  float w0 = 0.f, w1 = 0.f, w2 = 0.f, w3 = 0.f;
  int   d0 = 0, d1 = 0, d2 = 0, d3 = 0;
  int   e0 = 0, e1 = 0, e2 = 0, e3 = 0;

  // ---- prologue: fill the 4-slot ring
  if (t0 + 0 * grid < nTiles)
    issue_tile(emb, src, dst, w, nEdges, strideEdges, t0 + 0 * grid, eoff, sub, lds0, w0, d0, e0);
  if (t0 + 1 * grid < nTiles)
    issue_tile(emb, src, dst, w, nEdges, strideEdges, t0 + 1 * grid, eoff, sub, lds1, w1, d1, e1);
  if (t0 + 2 * grid < nTiles)
    issue_tile(emb, src, dst, w, nEdges, strideEdges, t0 + 2 * grid, eoff, sub, lds2, w2, d2, e2);
  if (t0 + 3 * grid < nTiles)
    issue_tile(emb, src, dst, w, nEdges, strideEdges, t0 + 3 * grid, eoff, sub, lds3, w3, d3, e3);

  // ---- steady state: wait-oldest, consume, refill same slot (4 tiles/iter)
  for (int t = t0; t < nTiles; t += DEPTH * grid) {
    { // slot 0: tile t
      int rt = t + 4 * grid;
      if (rt < nTiles) asm volatile("s_wait_asynccnt 0x3" ::: "memory");
      else             asm volatile("s_wait_asynccnt 0x0" ::: "memory");
      process_slot(&sh[0][tid], w0, d0, e0, nEdges, sub, out);
      if (rt < nTiles)
        issue_tile(emb, src, dst, w, nEdges, strideEdges, rt, eoff, sub, lds0, w0, d0, e0);
    }
    if (t + 1 * grid < nTiles) { // slot 1: tile t + g
      int rt = t + 5 * grid;
      if (rt < nTiles) asm volatile("s_wait_asynccnt 0x3" ::: "memory");
      else             asm volatile("s_wait_asynccnt 0x0" ::: "memory");
      process_slot(&sh[1][tid], w1, d1, e1, nEdges, sub, out);
      if (rt < nTiles)
        issue_tile(emb, src, dst, w, nEdges, strideEdges, rt, eoff, sub, lds1, w1, d1, e1);
    }
    if (t + 2 * grid < nTiles) { // slot 2: tile t + 2g
      int rt = t + 6 * grid;
      if (rt < nTiles) asm volatile("s_wait_asynccnt 0x3" ::: "memory");
      else             asm volatile("s_wait_asynccnt 0x0" ::: "memory");
      process_slot(&sh[2][tid], w2, d2, e2, nEdges, sub, out);
      if (rt < nTiles)
        issue_tile(emb, src, dst, w, nEdges, strideEdges, rt, eoff, sub, lds2, w2, d2, e2);
    }
    if (t + 3 * grid < nTiles) { // slot 3: tile t + 3g
      int rt = t + 7 * grid;
      if (rt < nTiles) asm volatile("s_wait_asynccnt 0x3" ::: "memory");
      else             asm volatile("s_wait_asynccnt 0x0" ::: "memory");
      process_slot(&sh[3][tid], w3, d3, e3, nEdges, sub, out);
      if (rt < nTiles)
        issue_tile(emb, src, dst, w, nEdges, strideEdges, rt, eoff, sub, lds3, w3, d3, e3);
    }
  }
}

// -------------------------------------------------- row L2 normalize (wave32)
__global__ __launch_bounds__(256) void lgc_normalize_kernel(float* __restrict__ out, int nNodes) {
  int lane = threadIdx.x & 31;
  int wid  = threadIdx.x >> 5;
  int node = blockIdx.x * 8 + wid;
  if (node >= nNodes) return;

  float2* row = (float2*)(out + (uint64_t)node * EMB_DIM);
  float2  v   = row[lane];
  float   ss  = v.x * v.x + v.y * v.y;
  #pragma unroll
  for (int off = 16; off > 0; off >>= 1)
    ss += __shfl_xor(ss, off, 32);

  float inv = 1.0f / fmaxf(sqrtf(ss), 1e-12f);
  v.x *= inv; v.y *= inv;
  row[lane] = v;
}

// ------------------------------------------------------------------- launch
extern "C" void kernel_launch(void* const* d_in, const int* in_sizes, int n_in,
                              void* d_out, int out_size, void* d_ws, size_t ws_size,
                              hipStream_t stream) {
  const float* emb = (const float*)d_in[0];
  const float* w   = (const float*)d_in[1];
  const int*   src = (const int*)d_in[2];
  const int*   dst = (const int*)d_in[3];
  float*       out = (float*)d_out;

  const int nNodes = in_sizes[0] / EMB_DIM;
  const int nEdges = in_sizes[1];

  // 1) zero accumulator (d_out) every call (graph-replay safe)
  int n4 = nNodes * (EMB_DIM / 4);
  lgc_zero_kernel<<<(n4 + 255) / 256, 256, 0, stream>>>((float4*)out, n4);

  // 2) edge gather-multiply-scatter, async-LDS 4-deep pipelined
  int nTiles = (nEdges + EPT - 1) / EPT;
  int grid   = nTiles < 12288 ? nTiles : 12288;   // persistent blocks
  lgc_edge_kernel<<<grid, BLOCK, 0, stream>>>(emb, w, src, dst, out, nEdges, nTiles);

  // 3) row-wise L2 normalize in place
  int gridN = (nNodes + 7) / 8;
  lgc_normalize_kernel<<<gridN, 256, 0, stream>>>(out, nNodes);
}